// MultiAttention_76931454206374
// MI455X (gfx1250) — compile-verified
//
#include <hip/hip_runtime.h>

typedef float v2f __attribute__((ext_vector_type(2)));
typedef float v8f __attribute__((ext_vector_type(8)));

#define B_      1024
#define L_      200
#define D_      128
#define H_      8
#define NTILE   8      // 128 / 16 column tiles
#define KSTEP   32     // 128 / 4  k-steps for 16x16x4 fp32 WMMA
#define ROWTILES 13    // ceil(200/16)
#define WAVES   8

__device__ __forceinline__ float sigmoidf(float x) {
    return 1.0f / (1.0f + __expf(-x));
}

__global__ __launch_bounds__(256, 1)
void fused_multiattention_kernel(const float* __restrict__ seq,   // (B,L,D)
                                 const float* __restrict__ tgt,   // (B,D)
                                 const float* __restrict__ mem,   // (B,D)
                                 const float* __restrict__ Wk,    // (D,D)
                                 const float* __restrict__ Wq,    // (D,D)
                                 const float* __restrict__ Wv,    // (D,D)
                                 const float* __restrict__ W1,    // (4D,H)
                                 const float* __restrict__ b1,    // (H)
                                 const float* __restrict__ W2,    // (H,1)
                                 const float* __restrict__ b2,    // (1)
                                 float* __restrict__ out,         // (B,D)
                                 float* __restrict__ att_out)     // (B,L)
{
    __shared__ float w1_s[4 * D_ * H_];   // 4096 floats
    __shared__ float v_s[D_];
    __shared__ float m_s[D_];
    __shared__ float out_s[D_];

    const int b    = blockIdx.x;
    const int tid  = threadIdx.x;
    const int lane = tid & 31;
    const int wave = tid >> 5;

    // ------------- stage small operands to LDS -------------
    for (int i = tid; i < 4 * D_ * H_; i += 256) w1_s[i] = W1[i];
    if (tid < D_) {
        // v = target_items[b] @ Wq  (one output column per thread, coalesced Wq reads)
        float acc = 0.f;
        const float* t = tgt + (size_t)b * D_;
        #pragma unroll 8
        for (int k = 0; k < D_; ++k) acc = fmaf(t[k], Wq[k * D_ + tid], acc);
        v_s[tid]   = acc;
        m_s[tid]   = mem[(size_t)b * D_ + tid];
        out_s[tid] = 0.f;
    }
    __syncthreads();

    // uniform tiny vectors into registers
    float b1v[H_], w2v[H_];
    #pragma unroll
    for (int j = 0; j < H_; ++j) { b1v[j] = b1[j]; w2v[j] = W2[j]; }
    const float b2v = b2[0];

    const int col  = lane & 15;   // column within 16-wide tile / A row index
    const int half = lane >> 4;   // 0: rows 0..7 of C, 1: rows 8..15
    const int kk   = 2 * half;    // K sub-offset inside each 4-wide k-step

    for (int tile = wave; tile < ROWTILES; tile += WAVES) {
        const int r0     = tile * 16;
        const int grow   = r0 + col;          // row (l) within batch for A fragment
        const bool rvalid = (grow < L_);
        const float* Arow = seq + ((size_t)b * L_ + grow) * D_ + kk;

        // prefetch this wave's next tile (global_prefetch_b8)
        {
            const int tnext = tile + WAVES;
            if (tnext < ROWTILES) {
                const int g2 = tnext * 16 + col;
                if (g2 < L_)
                    __builtin_prefetch(seq + ((size_t)b * L_ + g2) * D_ + kk, 0, 1);
            }
        }

        // ---- load A tile (16x128) in 16x16x4 fp32 A-fragment layout, NT hint ----
        v2f afrag[KSTEP];
        #pragma unroll
        for (int k = 0; k < KSTEP; ++k) {
            if (rvalid) {
                afrag[k] = __builtin_nontemporal_load((const v2f*)(Arow + 4 * k));
            } else {
                v2f z; z.x = 0.f; z.y = 0.f;
                afrag[k] = z;   // zero-pad tail rows: e=0 -> mask=0, V=0
            }
        }

        // ---- GEMM1 (e = A @ Wk) fused with masked alpha @ W1 partials ----
        float hp[8][H_];
        #pragma unroll
        for (int r = 0; r < 8; ++r)
            #pragma unroll
            for (int j = 0; j < H_; ++j) hp[r][j] = 0.f;

        #pragma unroll 1
        for (int nt = 0; nt < NTILE; ++nt) {
            const int   N  = nt * 16 + col;
            const float vN = v_s[N];
            const float mN = m_s[N];

            v8f acc = {0.f, 0.f, 0.f, 0.f, 0.f, 0.f, 0.f, 0.f};
            #pragma unroll
            for (int k = 0; k < KSTEP; ++k) {
                const int krow = 4 * k + kk;
                v2f bfrag;
                bfrag.x = Wk[(size_t)krow * D_ + N];
                bfrag.y = Wk[(size_t)(krow + 1) * D_ + N];
                acc = __builtin_amdgcn_wmma_f32_16x16x4_f32(
                        false, afrag[k], false, bfrag, (short)0, acc, false, false);
            }

            // W1 slices for this lane's column N (blocks of the concat)
            float w1a[H_], w1b[H_], w1c[H_], w1d[H_];
            #pragma unroll
            for (int j = 0; j < H_; ++j) {
                w1a[j] = w1_s[(size_t)N * H_ + j];
                w1b[j] = w1_s[(size_t)(D_ + N) * H_ + j];
                w1c[j] = w1_s[(size_t)(2 * D_ + N) * H_ + j];
                w1d[j] = w1_s[(size_t)(3 * D_ + N) * H_ + j];
            }

            #pragma unroll
            for (int r = 0; r < 8; ++r) {
                const float e   = acc[r];
                const float msk = (e != 0.f) ? 1.f : 0.f;
                const float t1  = msk * (vN - e);
                const float t2  = msk * vN * e;
                const float t3  = msk * (mN - e);
                const float t4  = msk * mN * e;
                #pragma unroll
                for (int j = 0; j < H_; ++j)
                    hp[r][j] = fmaf(t1, w1a[j],
                               fmaf(t2, w1b[j],
                               fmaf(t3, w1c[j],
                               fmaf(t4, w1d[j], hp[r][j]))));
            }
        }

        // ---- reduce hp across the 16 lanes of each half (column sum) ----
        #pragma unroll
        for (int r = 0; r < 8; ++r)
            #pragma unroll
            for (int j = 0; j < H_; ++j) {
                float x = hp[r][j];
                x += __shfl_xor(x, 1, 32);
                x += __shfl_xor(x, 2, 32);
                x += __shfl_xor(x, 4, 32);
                x += __shfl_xor(x, 8, 32);
                hp[r][j] = x;
            }

        // ---- attention scalar per row of this lane's half ----
        float attr[8];
        #pragma unroll
        for (int r = 0; r < 8; ++r) {
            float s2 = b2v;
            #pragma unroll
            for (int j = 0; j < H_; ++j)
                s2 = fmaf(sigmoidf(hp[r][j] + b1v[j]), w2v[j], s2);
            attr[r] = sigmoidf(s2);
        }
        if (col == 0) {
            #pragma unroll
            for (int r = 0; r < 8; ++r) {
                const int l = r0 + r + 8 * half;
                if (l < L_) att_out[(size_t)b * L_ + l] = attr[r];
            }
        }

        // ---- GEMM2 (V = A @ Wv) contracted with att on the fly ----
        #pragma unroll 1
        for (int nt = 0; nt < NTILE; ++nt) {
            const int N = nt * 16 + col;

            v8f acc = {0.f, 0.f, 0.f, 0.f, 0.f, 0.f, 0.f, 0.f};
            #pragma unroll
            for (int k = 0; k < KSTEP; ++k) {
                const int krow = 4 * k + kk;
                v2f bfrag;
                bfrag.x = Wv[(size_t)krow * D_ + N];
                bfrag.y = Wv[(size_t)(krow + 1) * D_ + N];
                acc = __builtin_amdgcn_wmma_f32_16x16x4_f32(
                        false, afrag[k], false, bfrag, (short)0, acc, false, false);
            }

            float s = 0.f;
            #pragma unroll
            for (int r = 0; r < 8; ++r) s = fmaf(attr[r], acc[r], s);
            atomicAdd(&out_s[N], s);   // ds_add_f32; lanes l and l+16 share column N
        }
    }

    __syncthreads();
    if (tid < D_) out[(size_t)b * D_ + tid] = out_s[tid];
}

extern "C" void kernel_launch(void* const* d_in, const int* in_sizes, int n_in,
                              void* d_out, int out_size, void* d_ws, size_t ws_size,
                              hipStream_t stream) {
    (void)in_sizes; (void)n_in; (void)out_size; (void)d_ws; (void)ws_size;
    const float* seq = (const float*)d_in[0];
    const float* tgt = (const float*)d_in[1];
    const float* mm  = (const float*)d_in[2];
    const float* Wk  = (const float*)d_in[3];
    const float* Wq  = (const float*)d_in[4];
    const float* Wv  = (const float*)d_in[5];
    const float* W1  = (const float*)d_in[6];
    const float* b1  = (const float*)d_in[7];
    const float* W2  = (const float*)d_in[8];
    const float* b2  = (const float*)d_in[9];

    float* out = (float*)d_out;           // (B,D) first
    float* att = out + (size_t)B_ * D_;   // then (B,L,1)

    fused_multiattention_kernel<<<B_, 256, 0, stream>>>(
        seq, tgt, mm, Wk, Wq, Wv, W1, b1, W2, b2, out, att);
}